// BasicGNNModel_81166291959925
// MI455X (gfx1250) — compile-verified
//
#include <hip/hip_runtime.h>

typedef __attribute__((ext_vector_type(2))) float v2f;
typedef __attribute__((ext_vector_type(8))) float v8f;

#define D_IN  128
#define D_HID 128
#define D_OUT 64

// ---------------- degree / norm precompute ----------------

__global__ void k_init_deg(float* deg, int n) {
    int i = blockIdx.x * blockDim.x + threadIdx.x;
    if (i < n) deg[i] = 1.0f;   // self-loop weight = 1
}

__global__ void k_accum_deg(const int* __restrict__ col, const float* __restrict__ w,
                            float* deg, int e) {
    int i = blockIdx.x * blockDim.x + threadIdx.x;
    if (i < e) unsafeAtomicAdd(&deg[col[i]], w[i]);
}

__global__ void k_dinv(const float* __restrict__ deg, float* dinv, float* dinv2, int n) {
    int i = blockIdx.x * blockDim.x + threadIdx.x;
    if (i < n) {
        float d = deg[i];
        float r = (d > 0.0f) ? rsqrtf(d) : 0.0f;
        dinv[i]  = r;
        dinv2[i] = r * r;   // self-loop norm
    }
}

__global__ void k_edge_norm(const int* __restrict__ row, const int* __restrict__ col,
                            const float* __restrict__ w, const float* __restrict__ dinv,
                            float* nrm, int e) {
    int i = blockIdx.x * blockDim.x + threadIdx.x;
    if (i < e) nrm[i] = dinv[row[i]] * w[i] * dinv[col[i]];
}

// ---------------- fp32 WMMA GEMM: lin = act(h) @ W ----------------
// One wave computes a 16-row stripe x (NT*16) columns, K-loop step 4.
// A 16x4 f32 layout: lanes0-15 M=lane {K0,K1}; lanes16-31 M=lane-16 {K2,K3}.
// B 4x16  f32 layout: v0: lanes0-15 K=0, lanes16-31 K=2; v1: K=1 / K=3, N=lane%16.
// C/D 16x16 f32: vgpr r -> M = 8*half + r, N = lane%16.

template <int DINc, int NT>
__global__ __launch_bounds__(256)
void k_gemm_wmma(const float* __restrict__ hin, const float* __restrict__ W,
                 float* __restrict__ lin, int n, int relu) {
    const int Dout = NT * 16;
    const int lane = threadIdx.x & 31;
    const int wave = threadIdx.x >> 5;
    const int half = lane >> 4;
    const int lidx = lane & 15;

    const long rowBase = ((long)blockIdx.x * 8 + wave) * 16;
    if (rowBase >= n) return;                       // wave-uniform exit: EXEC stays full

    long rowC = rowBase + lidx;                     // A-matrix M = lane%16 for both halves
    if (rowC > n - 1) rowC = n - 1;                 // clamp OOB loads (stores predicated)

    v8f acc[NT] = {};

    const float* aRow = hin + (size_t)rowC * DINc + 2 * half;

    for (int k0 = 0; k0 < DINc; k0 += 4) {
        float2 a2 = *(const float2*)(aRow + k0);
        if (relu) { a2.x = fmaxf(a2.x, 0.0f); a2.y = fmaxf(a2.y, 0.0f); }
        v2f av; av.x = a2.x; av.y = a2.y;

        const float* bBase = W + (size_t)(k0 + 2 * half) * Dout + lidx;
#pragma unroll
        for (int t = 0; t < NT; ++t) {
            v2f bv;
            bv.x = bBase[t * 16];
            bv.y = bBase[Dout + t * 16];
            acc[t] = __builtin_amdgcn_wmma_f32_16x16x4_f32(
                false, av, false, bv, (short)0, acc[t], false, false);
        }
    }

#pragma unroll
    for (int t = 0; t < NT; ++t) {
        const int c = t * 16 + lidx;
#pragma unroll
        for (int r = 0; r < 8; ++r) {
            long rr = rowBase + half * 8 + r;
            if (rr < n) lin[(size_t)rr * Dout + c] = acc[t][r];
        }
    }
}

// ---------------- out = bias + selfloop_norm * lin  (full overwrite) ----------------

template <int DoutT>
__global__ void k_init_out(const float* __restrict__ lin, const float* __restrict__ bias,
                           const float* __restrict__ dinv2, float* __restrict__ out, int n) {
    const int quads = DoutT / 4;
    int idx = blockIdx.x * blockDim.x + threadIdx.x;
    if (idx >= n * quads) return;
    int node = idx / quads;
    int c4   = (idx - node * quads) * 4;
    float  s = dinv2[node];
    float4 l = *(const float4*)(lin + (size_t)node * DoutT + c4);
    float4 b = *(const float4*)(bias + c4);
    float4 o;
    o.x = fmaf(s, l.x, b.x);
    o.y = fmaf(s, l.y, b.y);
    o.z = fmaf(s, l.z, b.z);
    o.w = fmaf(s, l.w, b.w);
    *(float4*)(out + (size_t)node * DoutT + c4) = o;
}

// ---------------- per-edge scatter: out[col] += norm * lin[row]  ----------------
// One wave per edge; lane handles VEC consecutive floats (Dout = 32*VEC).

template <int VEC>
__global__ __launch_bounds__(256)
void k_scatter(const int* __restrict__ row, const int* __restrict__ col,
               const float* __restrict__ nrm, const float* __restrict__ lin,
               float* __restrict__ out, int e) {
    const int waveId = blockIdx.x * 8 + (threadIdx.x >> 5);
    const int lane   = threadIdx.x & 31;
    if (waveId >= e) return;
    const int Dout = 32 * VEC;
    const int r = row[waveId];
    const int c = col[waveId];
    const float w = nrm[waveId];
    const float* src = lin + (size_t)r * Dout + lane * VEC;
    float*       dst = out + (size_t)c * Dout + lane * VEC;
    if (VEC == 4) {
        float4 v = *(const float4*)src;
        unsafeAtomicAdd(dst + 0, w * v.x);
        unsafeAtomicAdd(dst + 1, w * v.y);
        unsafeAtomicAdd(dst + 2, w * v.z);
        unsafeAtomicAdd(dst + 3, w * v.w);
    } else {
        float2 v = *(const float2*)src;
        unsafeAtomicAdd(dst + 0, w * v.x);
        unsafeAtomicAdd(dst + 1, w * v.y);
    }
}

// ---------------- launch ----------------

static inline int cdiv(int a, int b) { return (a + b - 1) / b; }

extern "C" void kernel_launch(void* const* d_in, const int* in_sizes, int n_in,
                              void* d_out, int out_size, void* d_ws, size_t ws_size,
                              hipStream_t stream) {
    const float* x   = (const float*)d_in[0];
    const int*   ei  = (const int*)d_in[1];
    const float* ew  = (const float*)d_in[2];
    const float* W1  = (const float*)d_in[3];
    const float* b1  = (const float*)d_in[4];
    const float* W2  = (const float*)d_in[5];
    const float* b2  = (const float*)d_in[6];
    const float* W3  = (const float*)d_in[7];
    const float* b3  = (const float*)d_in[8];
    float* out = (float*)d_out;

    const int N = in_sizes[0] / D_IN;
    const int E = in_sizes[1] / 2;
    const int* eRow = ei;         // sources
    const int* eCol = ei + E;     // targets

    // workspace layout (floats), all offsets 16B aligned
    float* ws    = (float*)d_ws;
    float* deg   = ws;
    float* dinv  = deg   + N;
    float* dinv2 = dinv  + N;
    float* nrm   = dinv2 + N;
    float* buf0  = nrm   + E;                 // aggregated features (N x 128)
    float* buf1  = buf0  + (size_t)N * D_HID; // linear-transform temp (N x 128)

    // --- normalization precompute (shared by all layers) ---
    k_init_deg <<<cdiv(N, 256), 256, 0, stream>>>(deg, N);
    k_accum_deg<<<cdiv(E, 256), 256, 0, stream>>>(eCol, ew, deg, E);
    k_dinv     <<<cdiv(N, 256), 256, 0, stream>>>(deg, dinv, dinv2, N);
    k_edge_norm<<<cdiv(E, 256), 256, 0, stream>>>(eRow, eCol, ew, dinv, nrm, E);

    const int gGemm = cdiv(N, 128);
    const int gScat = cdiv(E, 8);

    // --- layer 1: x @ W1 -> aggregate -> +b1 (relu deferred to next load) ---
    k_gemm_wmma<D_IN, 8><<<gGemm, 256, 0, stream>>>(x, W1, buf1, N, 0);
    k_init_out<128><<<cdiv(N * 32, 256), 256, 0, stream>>>(buf1, b1, dinv2, buf0, N);
    k_scatter<4><<<gScat, 256, 0, stream>>>(eRow, eCol, nrm, buf1, buf0, E);

    // --- layer 2: relu(h1) @ W2 -> aggregate -> +b2 ---
    k_gemm_wmma<D_HID, 8><<<gGemm, 256, 0, stream>>>(buf0, W2, buf1, N, 1);
    k_init_out<128><<<cdiv(N * 32, 256), 256, 0, stream>>>(buf1, b2, dinv2, buf0, N);
    k_scatter<4><<<gScat, 256, 0, stream>>>(eRow, eCol, nrm, buf1, buf0, E);

    // --- layer 3: relu(h2) @ W3 -> aggregate -> +b3, into d_out ---
    k_gemm_wmma<D_HID, 4><<<gGemm, 256, 0, stream>>>(buf0, W3, buf1, N, 1);
    k_init_out<64><<<cdiv(N * 16, 256), 256, 0, stream>>>(buf1, b3, dinv2, out, N);
    k_scatter<2><<<gScat, 256, 0, stream>>>(eRow, eCol, nrm, buf1, out, E);
}